// MemoryBank_25477746000174
// MI455X (gfx1250) — compile-verified
//
#include <hip/hip_runtime.h>

#define NQ      2048
#define NM      65536
#define DIM     512
#define KCH     128          // DIM / 4 (K-chunks per WMMA_16x16x4)
#define TOPK    16
#define NSEG    16
#define SEGROWS (NM / NSEG)  // 4096
#define WAVES   4
#define BLOCK   128
#define QT      32           // queries per workgroup (two 16-row M tiles)
#define BIGF    3.4e38f

typedef __attribute__((ext_vector_type(2))) float v2f;
typedef __attribute__((ext_vector_type(8))) float v8f;

// ---------------------------------------------------------------- norms ----
__global__ __launch_bounds__(256) void mb_norms(const float* __restrict__ query,
                                                const float* __restrict__ memory,
                                                float* __restrict__ qsq,
                                                float* __restrict__ msq) {
    int gwave = (blockIdx.x * blockDim.x + threadIdx.x) >> 5;
    int lane  = threadIdx.x & 31;
    if (gwave >= NM + NQ) return;
    const float* row = (gwave < NM) ? (memory + (size_t)gwave * DIM)
                                    : (query + (size_t)(gwave - NM) * DIM);
    float s = 0.f;
    for (int k = lane; k < DIM; k += 32) { float v = row[k]; s += v * v; }
    for (int off = 16; off > 0; off >>= 1) s += __shfl_xor(s, off);
    if (lane == 0) {
        if (gwave < NM) msq[gwave] = s;
        else            qsq[gwave - NM] = s;
    }
}

// ----------------------------------------------------------------- main ----
// grid: (NQ/QT, NSEG), block: 128 (4 waves). Each wave owns one 16-row
// B tile per iteration and multiplies it against TWO query M-tiles staged
// in LDS (shared B fragment -> half the L2 traffic, two independent WMMA
// accumulation chains -> pipelined matrix unit). Top-k kept on squared
// distances (sqrt deferred to the final merge).
__global__ __launch_bounds__(BLOCK) void mb_main(const float* __restrict__ query,
                                                 const float* __restrict__ memory,
                                                 const float* __restrict__ qsq,
                                                 const float* __restrict__ msq,
                                                 float* __restrict__ part_d,
                                                 int*   __restrict__ part_i) {
    __shared__ float qsA[2][KCH * 64];        // 64 KB: A fragments, 2 M-tiles
    __shared__ float qsq_s[QT];
    __shared__ float listd[WAVES][QT][TOPK];  // per-wave per-query top-16 (d^2)
    __shared__ int   listi[WAVES][QT][TOPK];
    __shared__ float curmax[WAVES][QT];
    __shared__ int   argmax[WAVES][QT];

    const int tid   = threadIdx.x;
    const int wave  = tid >> 5;
    const int lane  = tid & 31;
    const int half  = lane >> 4;
    const int l15   = lane & 15;
    const int qbase = blockIdx.x * QT;
    const int seg   = blockIdx.y;

    // Stage queries into LDS pre-swizzled to A-fragment layout:
    // A 16x4 f32: lane m holds K={0,1}, lane m+16 holds K={2,3}.
    for (int e = tid; e < 2 * KCH * 64; e += BLOCK) {
        int t  = e >> 13;            // which M tile (KCH*64 == 8192)
        int e2 = e & 8191;
        int kc = e2 >> 6, rem = e2 & 63, ln = rem >> 1, j = rem & 1;
        int m  = ln & 15;
        int k  = kc * 4 + ((ln >> 4) << 1) + j;
        qsA[t][e2] = query[(size_t)(qbase + t * 16 + m) * DIM + k];
    }
    if (tid < QT) qsq_s[tid] = qsq[qbase + tid];
    for (int e = tid; e < WAVES * QT * TOPK; e += BLOCK) {
        (&listd[0][0][0])[e] = BIGF;
        (&listi[0][0][0])[e] = 0;
    }
    // WAVES*QT == BLOCK, every thread initializes one slot
    (&curmax[0][0])[tid] = BIGF;
    (&argmax[0][0])[tid] = 0;
    __syncthreads();

    int idx = 0;  // candidate memory-row index for this lane (set per iter)

    auto try_insert = [&](float d2v, int tofs, int r) {
        int   q   = tofs + r + (half << 3);
        float thr = curmax[wave][q];
        unsigned long long mask = __ballot(d2v < thr);
        while (mask) {
            int l = (int)__builtin_ctzll(mask);
            mask &= mask - 1;
            float dl = __shfl(d2v, l);
            int   il = __shfl(idx, l);
            int   ql = tofs + r + ((l >> 4) << 3);
            if (lane == 0) {
                if (dl < curmax[wave][ql]) {
                    int am = argmax[wave][ql];
                    listd[wave][ql][am] = dl;
                    listi[wave][ql][am] = il;
                    float mx = -BIGF; int ai = 0;
                    for (int j2 = 0; j2 < TOPK; ++j2) {
                        float v = listd[wave][ql][j2];
                        if (v > mx) { mx = v; ai = j2; }
                    }
                    curmax[wave][ql] = mx;
                    argmax[wave][ql] = ai;
                }
            }
        }
    };

    const int iters = SEGROWS / (16 * WAVES);  // 64
    for (int it = 0; it < iters; ++it) {
        const int nbase = seg * SEGROWS + (it * WAVES + wave) * 16;
        // B 4x16 f32: V0 = {K=0 | K=2}, V1 = {K=1 | K=3} across lane halves.
        const float* brow = memory + (size_t)(nbase + l15) * DIM + half * 2;

        v8f acc0 = {0.f, 0.f, 0.f, 0.f, 0.f, 0.f, 0.f, 0.f};
        v8f acc1 = {0.f, 0.f, 0.f, 0.f, 0.f, 0.f, 0.f, 0.f};
        #pragma unroll 4
        for (int kc = 0; kc < KCH; ++kc) {
            v2f b  = *(const v2f*)(brow + kc * 4);
            v2f a0 = *(const v2f*)(&qsA[0][kc * 64 + lane * 2]);
            v2f a1 = *(const v2f*)(&qsA[1][kc * 64 + lane * 2]);
            acc0 = __builtin_amdgcn_wmma_f32_16x16x4_f32(
                false, a0, false, b, (short)0, acc0, false, false);
            acc1 = __builtin_amdgcn_wmma_f32_16x16x4_f32(
                false, a1, false, b, (short)0, acc1, false, false);
        }

        const float msq_l = msq[nbase + l15];
        idx = nbase + l15;

        #pragma unroll
        for (int r = 0; r < 8; ++r) {
            // C/D layout: VGPR r, lanes 0-15 -> (M=r, N=lane); 16-31 -> (M=r+8).
            float d0 = qsq_s[r + (half << 3)]      + msq_l - 2.f * acc0[r];
            float d1 = qsq_s[16 + r + (half << 3)] + msq_l - 2.f * acc1[r];
            d0 = d0 > 0.f ? d0 : 0.f;
            d1 = d1 > 0.f ? d1 : 0.f;
            try_insert(d0, 0, r);
            try_insert(d1, 16, r);
        }
    }
    __syncthreads();

    // Merge the 4 wave lists per query -> one partial top-16 per (seg, query).
    if (tid < QT) {
        const int q = tid;
        float bestd[TOPK]; int besti[TOPK];
        for (int j = 0; j < TOPK; ++j) { bestd[j] = BIGF; besti[j] = 0; }
        float mx = BIGF; int am = 0;
        for (int w = 0; w < WAVES; ++w)
            for (int j = 0; j < TOPK; ++j) {
                float v = listd[w][q][j];
                if (v < mx) {
                    bestd[am] = v; besti[am] = listi[w][q][j];
                    mx = -BIGF;
                    for (int j2 = 0; j2 < TOPK; ++j2)
                        if (bestd[j2] > mx) { mx = bestd[j2]; am = j2; }
                }
            }
        size_t off = ((size_t)seg * NQ + (qbase + q)) * TOPK;
        for (int j = 0; j < TOPK; ++j) { part_d[off + j] = bestd[j]; part_i[off + j] = besti[j]; }
    }
}

// ---------------------------------------------------------------- final ----
__global__ __launch_bounds__(256) void mb_final(const float* __restrict__ part_d,
                                                const int*   __restrict__ part_i,
                                                float* __restrict__ out_d,
                                                float* __restrict__ out_i) {
    int q = blockIdx.x * blockDim.x + threadIdx.x;
    if (q >= NQ) return;
    float bestd[TOPK]; int besti[TOPK];
    for (int j = 0; j < TOPK; ++j) { bestd[j] = BIGF; besti[j] = 0; }
    float mx = BIGF; int am = 0;
    for (int s = 0; s < NSEG; ++s) {
        size_t off = ((size_t)s * NQ + q) * TOPK;
        for (int j = 0; j < TOPK; ++j) {
            float v = part_d[off + j];
            if (v < mx) {
                bestd[am] = v; besti[am] = part_i[off + j];
                mx = -BIGF;
                for (int j2 = 0; j2 < TOPK; ++j2)
                    if (bestd[j2] > mx) { mx = bestd[j2]; am = j2; }
            }
        }
    }
    // Sort ascending on squared distance (same order as distance).
    for (int i = 1; i < TOPK; ++i) {
        float dv = bestd[i]; int iv = besti[i]; int j = i - 1;
        while (j >= 0 && bestd[j] > dv) {
            bestd[j + 1] = bestd[j]; besti[j + 1] = besti[j]; --j;
        }
        bestd[j + 1] = dv; besti[j + 1] = iv;
    }
    for (int j = 0; j < TOPK; ++j) {
        out_d[(size_t)q * TOPK + j] = __builtin_sqrtf(bestd[j]);
        out_i[(size_t)q * TOPK + j] = (float)besti[j];
    }
}

// --------------------------------------------------------------- launch ----
extern "C" void kernel_launch(void* const* d_in, const int* in_sizes, int n_in,
                              void* d_out, int out_size, void* d_ws, size_t ws_size,
                              hipStream_t stream) {
    const float* query  = (const float*)d_in[0];
    const float* memory = (const float*)d_in[1];
    // d_in[2] is k (== 16), baked in at compile time.

    float* ws     = (float*)d_ws;
    float* qsq    = ws;                     // 2048
    float* msq    = ws + NQ;                // 65536
    float* part_d = msq + NM;               // NSEG*NQ*TOPK = 524288
    int*   part_i = (int*)(part_d + (size_t)NSEG * NQ * TOPK);

    float* out_d = (float*)d_out;              // 2048*16 distances
    float* out_i = out_d + (size_t)NQ * TOPK;  // 2048*16 indices (as float)

    {
        int waves  = NM + NQ;
        int blocks = (waves * 32 + 255) / 256;
        mb_norms<<<blocks, 256, 0, stream>>>(query, memory, qsq, msq);
    }
    {
        dim3 grid(NQ / QT, NSEG);
        mb_main<<<grid, BLOCK, 0, stream>>>(query, memory, qsq, msq, part_d, part_i);
    }
    {
        mb_final<<<(NQ + 255) / 256, 256, 0, stream>>>(part_d, part_i, out_d, out_i);
    }
}